// BotGCN_Single_80573586473699
// MI455X (gfx1250) — compile-verified
//
#include <hip/hip_runtime.h>
#include <hip/hip_bf16.h>

#define N_NODES 50000
#define E_EDGES 800000
#define EMB 384

typedef __attribute__((ext_vector_type(2))) float v2f;
typedef __attribute__((ext_vector_type(8))) float v8f;

__device__ __forceinline__ float leaky01(float x) {
    return x > 0.0f ? x : 0.01f * x;
}

// ---------------------------------------------------------------------------
// WMMA fp32 GEMM body: one wave computes a (16*MT) x (16*NT) output tile.
// A-frag layout (ISA 7.12.2, f32 16x4): lanes 0-15 -> K=0,1 ; lanes 16-31 -> K=2,3
// B-frag layout (4x16):                 lanes 0-15 -> K=0,1 ; lanes 16-31 -> K=2,3
// C/D layout: vgpr v, lane<16 -> M=v ; lane>=16 -> M=v+8 ; N = lane&15
// MT=2 reuses each B fragment across two A fragments (halves B traffic).
// All pointers are strength-reduced to per-iteration increments (no 64-bit
// multiplies inside the k-loop).
// ---------------------------------------------------------------------------
template <bool ACT, int NT, int MT>
__device__ __forceinline__ void gemm_body(const float* __restrict__ A, int lda,
                                          const float* __restrict__ B, int ldb,
                                          const float* __restrict__ bias,
                                          float* __restrict__ C, int ldc, int K,
                                          int tileM, int tileN, int lo, int hi) {
    v8f acc[MT][NT];
    v8f zero = {};
#pragma unroll
    for (int mi = 0; mi < MT; ++mi)
#pragma unroll
        for (int t = 0; t < NT; ++t) acc[mi][t] = zero;

    const float* aPtr[MT];
#pragma unroll
    for (int mi = 0; mi < MT; ++mi)
        aPtr[mi] = A + (size_t)(tileM + mi * 16 + lo) * lda + 2 * hi;

    // B pointer: row (2*hi) of the current K-block; advance by 4 rows per step.
    const float* bPtr = B + (size_t)(2 * hi) * ldb + tileN + lo;
    const size_t bStep = 4 * (size_t)ldb;

#pragma unroll 2
    for (int kb = 0; kb < K; kb += 4) {
        v2f a[MT];
#pragma unroll
        for (int mi = 0; mi < MT; ++mi) {
            a[mi] = *(const v2f*)(aPtr[mi]);  // 8B-aligned global_load_b64
            aPtr[mi] += 4;
        }
#pragma unroll
        for (int t = 0; t < NT; ++t) {
            v2f b;
            b.x = bPtr[t * 16];        // K = kb + 2*hi
            b.y = bPtr[ldb + t * 16];  // K = kb + 2*hi + 1
#pragma unroll
            for (int mi = 0; mi < MT; ++mi)
                acc[mi][t] = __builtin_amdgcn_wmma_f32_16x16x4_f32(
                    false, a[mi], false, b, (short)0, acc[mi][t], false, false);
        }
        bPtr += bStep;
    }

#pragma unroll
    for (int mi = 0; mi < MT; ++mi)
#pragma unroll
        for (int t = 0; t < NT; ++t) {
            const int col = tileN + t * 16 + lo;
            const float bv = bias ? bias[col] : 0.0f;
#pragma unroll
            for (int v = 0; v < 8; ++v) {
                const int row = tileM + mi * 16 + v + 8 * hi;
                float x = acc[mi][t][v] + bv;
                if (ACT) x = leaky01(x);
                C[(size_t)row * ldc + col] = x;
            }
        }
}

template <bool ACT, int NT>
__global__ void gemm_wmma(const float* __restrict__ A, int lda,
                          const float* __restrict__ B, int ldb,
                          const float* __restrict__ bias,
                          float* __restrict__ C, int ldc, int K, int M) {
    const int lane = threadIdx.x & 31;
    const int lo = lane & 15;
    const int hi = lane >> 4;
    const int tileN = blockIdx.x * (16 * NT);
    const int tileM = blockIdx.y * 32;  // two 16-row tiles per wave (when in range)

    if (tileM + 32 <= M) {
        gemm_body<ACT, NT, 2>(A, lda, B, ldb, bias, C, ldc, K, tileM, tileN, lo, hi);
    } else {
        gemm_body<ACT, NT, 1>(A, lda, B, ldb, bias, C, ldc, K, tileM, tileN, lo, hi);
    }
}

// num_prop (K=4) and cat_prop (K=3) projections to 128 cols each (VALU; K too
// small for WMMA). Writes x0[:,128:256] and x0[:,256:384].
__global__ void numcat_kernel(const float* __restrict__ nump,
                              const float* __restrict__ catp,
                              const float* __restrict__ Wn, const float* __restrict__ bn,
                              const float* __restrict__ Wc, const float* __restrict__ bc,
                              float* __restrict__ x0) {
    int idx = blockIdx.x * blockDim.x + threadIdx.x;
    if (idx >= N_NODES * 128) return;
    int node = idx >> 7;
    int c = idx & 127;
    const float* np = nump + node * 4;
    float s = bn[c];
#pragma unroll
    for (int k = 0; k < 4; ++k) s += np[k] * Wn[k * 128 + c];
    const float* cp = catp + node * 3;
    float s2 = bc[c];
#pragma unroll
    for (int k = 0; k < 3; ++k) s2 += cp[k] * Wc[k * 128 + c];
    x0[(size_t)node * EMB + 128 + c] = leaky01(s);
    x0[(size_t)node * EMB + 256 + c] = leaky01(s2);
}

__global__ void zero_kernel(float* __restrict__ p, int n) {
    int i = blockIdx.x * blockDim.x + threadIdx.x;
    if (i < n) p[i] = 0.0f;
}

__global__ void degree_kernel(const long long* __restrict__ ei, float* __restrict__ deg) {
    int e = blockIdx.x * blockDim.x + threadIdx.x;
    if (e < E_EDGES) {
        long long d = ei[E_EDGES + e];  // dst row
        unsafeAtomicAdd(&deg[d], 1.0f);
    }
}

__global__ void dinv_kernel(const float* __restrict__ deg, float* __restrict__ dinv, int n) {
    int i = blockIdx.x * blockDim.x + threadIdx.x;
    if (i < n) dinv[i] = rsqrtf(deg[i] + 1.0f);  // +1 self loop
}

// agg = xw * dinv^2 + bias  (self-loop contribution + conv bias)
__global__ void selfloop_kernel(const float* __restrict__ xw,
                                const float* __restrict__ dinv,
                                const float* __restrict__ bias,
                                float* __restrict__ agg) {
    int tid = blockIdx.x * blockDim.x + threadIdx.x;
    if (tid >= N_NODES * EMB) return;
    int node = tid / EMB;
    int c = tid - node * EMB;
    float di = dinv[node];
    agg[tid] = xw[tid] * di * di + bias[c];
}

// agg[dst] += xw[src] * dinv[src] * dinv[dst]   (one wave per edge)
__global__ void scatter_kernel(const long long* __restrict__ ei,
                               const float* __restrict__ xw,
                               const float* __restrict__ dinv,
                               float* __restrict__ agg) {
    int wave = threadIdx.x >> 5;
    int lane = threadIdx.x & 31;
    int e = blockIdx.x * (blockDim.x >> 5) + wave;
    if (e >= E_EDGES) return;
    long long s = ei[e];
    long long d = ei[E_EDGES + e];
    float nrm = dinv[s] * dinv[d];
    const float* xs = xw + (size_t)s * EMB;
    float* ad = agg + (size_t)d * EMB;
#pragma unroll
    for (int c = lane; c < EMB; c += 32)
        unsafeAtomicAdd(&ad[c], xs[c] * nrm);
}

// out[node, 0:2] = x[node,:] @ W_o2[384x2] + b_o2   (one wave per node)
__global__ void out_kernel(const float* __restrict__ x, const float* __restrict__ W,
                           const float* __restrict__ b, float* __restrict__ out) {
    int wave = threadIdx.x >> 5;
    int lane = threadIdx.x & 31;
    int node = blockIdx.x * (blockDim.x >> 5) + wave;
    if (node >= N_NODES) return;
    const float* xr = x + (size_t)node * EMB;
    float s0 = 0.0f, s1 = 0.0f;
    for (int k = lane; k < EMB; k += 32) {
        float xv = xr[k];
        s0 += xv * W[k * 2 + 0];
        s1 += xv * W[k * 2 + 1];
    }
#pragma unroll
    for (int m = 16; m >= 1; m >>= 1) {
        s0 += __shfl_xor(s0, m, 32);
        s1 += __shfl_xor(s1, m, 32);
    }
    if (lane == 0) {
        out[node * 2 + 0] = s0 + b[0];
        out[node * 2 + 1] = s1 + b[1];
    }
}

extern "C" void kernel_launch(void* const* d_in, const int* in_sizes, int n_in,
                              void* d_out, int out_size, void* d_ws, size_t ws_size,
                              hipStream_t stream) {
    const float* des  = (const float*)d_in[0];
    // d_in[1] = tweet (unused by reference forward pass)
    const float* nump = (const float*)d_in[2];
    const float* catp = (const float*)d_in[3];
    const long long* ei = (const long long*)d_in[4];  // int64 [2, E]
    const float *W_des = (const float*)d_in[5],  *b_des = (const float*)d_in[6];
    const float *W_num = (const float*)d_in[7],  *b_num = (const float*)d_in[8];
    const float *W_cat = (const float*)d_in[9],  *b_cat = (const float*)d_in[10];
    const float *W_in  = (const float*)d_in[11], *b_in  = (const float*)d_in[12];
    const float *W_g1  = (const float*)d_in[13], *b_g1  = (const float*)d_in[14];
    const float *W_g2  = (const float*)d_in[15], *b_g2  = (const float*)d_in[16];
    const float *W_o1  = (const float*)d_in[17], *b_o1  = (const float*)d_in[18];
    const float *W_o2  = (const float*)d_in[19], *b_o2  = (const float*)d_in[20];
    float* out = (float*)d_out;

    const size_t NE = (size_t)N_NODES * EMB;
    float* bufA = (float*)d_ws;      // x buffers, 76.8 MB each
    float* bufB = bufA + NE;
    float* bufC = bufB + NE;
    float* deg  = bufC + NE;
    float* dinv = deg + N_NODES;

    const int GY = (N_NODES + 31) / 32;  // 1563 row blocks (32 rows per wave)
    dim3 blk32(32);

    // x0[:,0:128] = leaky(des @ W_des + b_des)   (M=50000, K=768, N=128)
    gemm_wmma<true, 4><<<dim3(128 / 64, GY), blk32, 0, stream>>>(
        des, 768, W_des, 128, b_des, bufA, EMB, 768, N_NODES);
    // x0[:,128:384] = leaky(num/cat projections)
    numcat_kernel<<<(N_NODES * 128 + 255) / 256, 256, 0, stream>>>(
        nump, catp, W_num, b_num, W_cat, b_cat, bufA);

    // x1 = leaky(x0 @ W_in + b_in)
    gemm_wmma<true, 4><<<dim3(EMB / 64, GY), blk32, 0, stream>>>(
        bufA, EMB, W_in, EMB, b_in, bufB, EMB, EMB, N_NODES);

    // degree / dinv (shared by both convs)
    zero_kernel<<<(N_NODES + 255) / 256, 256, 0, stream>>>(deg, N_NODES);
    degree_kernel<<<(E_EDGES + 255) / 256, 256, 0, stream>>>(ei, deg);
    dinv_kernel<<<(N_NODES + 255) / 256, 256, 0, stream>>>(deg, dinv, N_NODES);

    // conv1: xw = x1 @ W_g1 ; agg = scatter(xw*norm) + xw*dinv^2 + b_g1
    gemm_wmma<false, 4><<<dim3(EMB / 64, GY), blk32, 0, stream>>>(
        bufB, EMB, W_g1, EMB, nullptr, bufC, EMB, EMB, N_NODES);
    selfloop_kernel<<<(N_NODES * EMB + 255) / 256, 256, 0, stream>>>(bufC, dinv, b_g1, bufA);
    scatter_kernel<<<(E_EDGES + 7) / 8, 256, 0, stream>>>(ei, bufC, dinv, bufA);

    // conv2
    gemm_wmma<false, 4><<<dim3(EMB / 64, GY), blk32, 0, stream>>>(
        bufA, EMB, W_g2, EMB, nullptr, bufB, EMB, EMB, N_NODES);
    selfloop_kernel<<<(N_NODES * EMB + 255) / 256, 256, 0, stream>>>(bufB, dinv, b_g2, bufC);
    scatter_kernel<<<(E_EDGES + 7) / 8, 256, 0, stream>>>(ei, bufB, dinv, bufC);

    // head: x4 = leaky(x3 @ W_o1 + b_o1) ; out = x4 @ W_o2 + b_o2
    gemm_wmma<true, 4><<<dim3(EMB / 64, GY), blk32, 0, stream>>>(
        bufC, EMB, W_o1, EMB, b_o1, bufA, EMB, EMB, N_NODES);
    out_kernel<<<(N_NODES + 7) / 8, 256, 0, stream>>>(bufA, W_o2, b_o2, out);
}